// HyperbolicLinearHyperboloidFHNN_18743237279914
// MI455X (gfx1250) — compile-verified
//
#include <hip/hip_runtime.h>
#include <math.h>

typedef __attribute__((ext_vector_type(2))) float v2f;
typedef __attribute__((ext_vector_type(8))) float v8f;

// Padded LDS row stride (floats): 68 keeps float4 staging stores 16B-aligned
// (68*4 = 272 ≡ 0 mod 16) and makes B-fragment ds_load_b64 conflict-free
// (bank = (4m + c) % 64, distinct for m = 0..15).
#define WSTRIDE 68

__global__ __launch_bounds__(256)
void hyp_linear_wmma_f32(const float* __restrict__ x,
                         const float* __restrict__ w,
                         const float* __restrict__ bias,
                         const float* __restrict__ scale,
                         float* __restrict__ out)
{
    __shared__ float wlds[64 * WSTRIDE];

    const int tid  = threadIdx.x;
    const int wave = tid >> 5;
    const int lane = tid & 31;
    const int half = lane >> 4;   // 0 = lanes 0-15, 1 = lanes 16-31
    const int m    = lane & 15;   // A: row-in-tile; B/C: col-in-tile

    // ---- stage W (64x64 f32, row-major [o][i]) into padded LDS (coalesced float4)
    {
        const float4* w4 = (const float4*)w;
        #pragma unroll
        for (int j = 0; j < 4; ++j) {
            int linear = tid + j * 256;      // 1024 float4 total
            int o  = linear >> 4;            // 16 float4 per 64-float row
            int i4 = (linear & 15) << 2;
            float4 v = w4[linear];
            *(float4*)&wlds[o * WSTRIDE + i4] = v;
        }
    }
    __syncthreads();

    const long rowBase = (long)blockIdx.x * 128 + (long)wave * 16;

    // 16x16 f32 accumulators, 4 N-tiles covering OUT_DIM = 64
    v8f c0 = {}, c1 = {}, c2 = {}, c3 = {};

    // A fragment source: lane (half,m) reads x[rowBase+m][k0 + 2*half .. +1]
    const float* aptr = x + (rowBase + m) * 64 + 2 * half;
    // B fragment source: lane (half,m) reads W[n = m+16*nt][k0 + 2*half .. +1]
    const float* b0 = &wlds[(m +  0) * WSTRIDE + 2 * half];
    const float* b1 = &wlds[(m + 16) * WSTRIDE + 2 * half];
    const float* b2 = &wlds[(m + 32) * WSTRIDE + 2 * half];
    const float* b3 = &wlds[(m + 48) * WSTRIDE + 2 * half];

    #pragma unroll
    for (int k0 = 0; k0 < 64; k0 += 4) {
        v2f a   = *(const v2f*)(aptr + k0);
        v2f fb0 = *(const v2f*)(b0 + k0);
        v2f fb1 = *(const v2f*)(b1 + k0);
        v2f fb2 = *(const v2f*)(b2 + k0);
        v2f fb3 = *(const v2f*)(b3 + k0);
        c0 = __builtin_amdgcn_wmma_f32_16x16x4_f32(false, a, false, fb0, (short)0, c0, false, false);
        c1 = __builtin_amdgcn_wmma_f32_16x16x4_f32(false, a, false, fb1, (short)0, c1, false, false);
        c2 = __builtin_amdgcn_wmma_f32_16x16x4_f32(false, a, false, fb2, (short)0, c2, false, false);
        c3 = __builtin_amdgcn_wmma_f32_16x16x4_f32(false, a, false, fb3, (short)0, c3, false, false);
    }

    // ---- bias add (broadcast over rows)
    const float bb0 = bias[m];
    const float bb1 = bias[m + 16];
    const float bb2 = bias[m + 32];
    const float bb3 = bias[m + 48];
    #pragma unroll
    for (int r = 0; r < 8; ++r) {
        c0[r] += bb0; c1[r] += bb1; c2[r] += bb2; c3[r] += bb3;
    }

    // ---- hyperboloid epilogue (C = 1.0: inv_sqrt_c = 1, 1/C = 1)
    const float escale  = expf(scale[0]);
    const int   srcLane = lane & 16;   // broadcast source (col 0 of each half)

    #pragma unroll
    for (int r = 0; r < 8; ++r) {
        float v0 = c0[r], v1 = c1[r], v2 = c2[r], v3 = c3[r];

        // norm^2 over cols 1..63: exclude col 0 (tile 0, lane m==0)
        float sq = v1 * v1 + v2 * v2 + v3 * v3 + ((m == 0) ? 0.0f : v0 * v0);
        sq += __shfl_xor(sq, 1, 32);   // xor masks 1,2,4,8 stay within each
        sq += __shfl_xor(sq, 2, 32);   // 16-lane half (one row per half)
        sq += __shfl_xor(sq, 4, 32);
        sq += __shfl_xor(sq, 8, 32);

        float x0   = __shfl(v0, srcLane, 32);              // h[row][0]
        float res0 = escale / (1.0f + expf(-x0)) + 1.0f + 1e-5f;
        float s    = sqrtf(res0 * res0 - 1.0f) * rsqrtf(sq);

        float* orow = out + (rowBase + r + 8 * half) * 64 + m;
        orow[0]  = (m == 0) ? res0 : s * v0;
        orow[16] = s * v1;
        orow[32] = s * v2;
        orow[48] = s * v3;
    }
}

extern "C" void kernel_launch(void* const* d_in, const int* in_sizes, int n_in,
                              void* d_out, int out_size, void* d_ws, size_t ws_size,
                              hipStream_t stream) {
    (void)n_in; (void)out_size; (void)d_ws; (void)ws_size;
    const float* x     = (const float*)d_in[0];
    const float* w     = (const float*)d_in[1];
    const float* bias  = (const float*)d_in[2];
    const float* scale = (const float*)d_in[3];
    float* out = (float*)d_out;

    const long rows = (long)in_sizes[0] / 64;   // B = 2,097,152
    const int  grid = (int)((rows + 127) / 128);

    hipLaunchKernelGGL(hyp_linear_wmma_f32, dim3(grid), dim3(256), 0, stream,
                       x, w, bias, scale, out);
}